// GroupedLinear_26680336843599
// MI455X (gfx1250) — compile-verified
//
#include <hip/hip_runtime.h>

// Grouped GEMM: out[l] = x[l] @ w[l] + b[l]
//   L=64, T=8192, DIN=128, DOUT=128, all fp32.
// Bandwidth-bound (~540MB @ 23.3TB/s ~= 23us floor); use fp32 WMMA
// (V_WMMA_F32_16X16X4_F32) so compute stays off the critical path with
// no precision loss.

typedef float v2f __attribute__((ext_vector_type(2)));
typedef float v8f __attribute__((ext_vector_type(8)));

#define GL_L    64
#define GL_T    8192
#define GL_DIN  128
#define GL_DOUT 128

// Row padding for LDS-resident W: need 2*stride % 64 == 16 (banks) so the two
// half-waves of a B-fragment read hit disjoint bank sets. 136 = 128 + 8.
#define WSTRIDE 136

#define ROWS_PER_BLOCK 128            // 8 waves x 16 rows
#define BLOCKS_PER_LAYER (GL_T / ROWS_PER_BLOCK)   // 64

__global__ __launch_bounds__(256) void grouped_linear_wmma_f32(
    const float* __restrict__ x,   // [L, T, DIN]
    const float* __restrict__ w,   // [L, DIN, DOUT]
    const float* __restrict__ b,   // [L, 1, DOUT]
    float* __restrict__ out)       // [L, T, DOUT]
{
    __shared__ float w_lds[GL_DIN * WSTRIDE];   // padded rows of w[layer]
    __shared__ float b_lds[GL_DOUT];

    const int layer  = blockIdx.x / BLOCKS_PER_LAYER;
    const int rowblk = blockIdx.x % BLOCKS_PER_LAYER;
    const int tid    = threadIdx.x;
    const int wave   = tid >> 5;
    const int lane   = tid & 31;

    // ---- cooperative load of w[layer] into padded LDS (coalesced float4) ----
    {
        const float* wsrc = w + (size_t)layer * GL_DIN * GL_DOUT;
        #pragma unroll
        for (int i = tid; i < (GL_DIN * GL_DOUT) / 4; i += 256) {
            const int row = (i * 4) / GL_DOUT;      // DOUT % 4 == 0 -> stays in-row
            const int col = (i * 4) % GL_DOUT;
            const float4 v = *(const float4*)(wsrc + row * GL_DOUT + col);
            *(float4*)&w_lds[row * WSTRIDE + col] = v;
        }
        if (tid < GL_DOUT)
            b_lds[tid] = b[(size_t)layer * GL_DOUT + tid];
    }
    __syncthreads();

    // ---- A fragments: 16 rows x 128 K, ISA layout for V_WMMA_F32_16X16X4_F32
    //      lane l: M = l%16, VGPR j holds K = kstep*4 + 2*(l/16) + j  -> v2f ----
    const int mrow  = lane & 15;          // M within tile
    const int khalf = lane >> 4;          // 0 or 1
    const int m0    = rowblk * ROWS_PER_BLOCK + wave * 16;

    const float* xrow = x + (((size_t)layer * GL_T) + (m0 + mrow)) * GL_DIN + 2 * khalf;

    v2f areg[32];
    #pragma unroll
    for (int ks = 0; ks < 32; ++ks)
        areg[ks] = *(const v2f*)(xrow + ks * 4);   // 8B aligned, K-sequential

    // ---- 8 N-tiles of 16 columns each; B frag from LDS, accumulate 32 WMMAs ----
    #pragma unroll
    for (int nt = 0; nt < 8; ++nt) {
        const int ncol = nt * 16 + mrow;           // N = lane%16 within tile
        const float* bptr = &w_lds[(2 * khalf) * WSTRIDE + ncol];

        v8f acc = {0.f, 0.f, 0.f, 0.f, 0.f, 0.f, 0.f, 0.f};

        #pragma unroll
        for (int ks = 0; ks < 32; ++ks) {
            v2f bfrag;
            bfrag.x = bptr[(ks * 4 + 0) * WSTRIDE];   // K = 4ks + 2h
            bfrag.y = bptr[(ks * 4 + 1) * WSTRIDE];   // K = 4ks + 2h + 1
            acc = __builtin_amdgcn_wmma_f32_16x16x4_f32(
                /*neg_a=*/false, areg[ks],
                /*neg_b=*/false, bfrag,
                /*c_mod=*/(short)0, acc,
                /*reuse_a=*/false, /*reuse_b=*/false);
        }

        // C/D layout: VGPR r -> row m0 + r + 8*khalf, col = ncol
        const float bias = b_lds[ncol];
        float* orow = out + (((size_t)layer * GL_T) + (size_t)(m0 + 8 * khalf)) * GL_DOUT + ncol;
        #pragma unroll
        for (int r = 0; r < 8; ++r)
            orow[(size_t)r * GL_DOUT] = acc[r] + bias;
    }
}

extern "C" void kernel_launch(void* const* d_in, const int* in_sizes, int n_in,
                              void* d_out, int out_size, void* d_ws, size_t ws_size,
                              hipStream_t stream) {
    const float* x = (const float*)d_in[0];
    const float* w = (const float*)d_in[1];
    const float* b = (const float*)d_in[2];
    float* out = (float*)d_out;

    const dim3 grid(GL_L * BLOCKS_PER_LAYER);   // 4096 blocks
    const dim3 block(256);                      // 8 wave32s
    grouped_linear_wmma_f32<<<grid, block, 0, stream>>>(x, w, b, out);
}